// ToposYonedaModel_9783935500328
// MI455X (gfx1250) — compile-verified
//
#include <hip/hip_runtime.h>
#include <math.h>

// score[b] = sigmoid(logits[u[b]*V + v[b]]),  V = 8192, B = 16384.
// Latency-bound two-level gather:
//   - u[i], v[i]: coalesced global_load_b32
//   - logits gather: scattered global_load_b32 (saddr + scaled vgpr offset),
//     32 independent loads in flight per wave32
//   - sigmoid: v_mul (×-log2e) -> v_exp_f32 -> v_add -> v_rcp_f32
//     (hardware rcp is accuracy-matched to the hardware exp approximation;
//      avoids the full IEEE v_div_scale/div_fmas/div_fixup sequence)
__global__ __launch_bounds__(256) void topos_yoneda_gather_sigmoid(
    const int* __restrict__ u,
    const int* __restrict__ v,
    const float* __restrict__ logits,
    float* __restrict__ out,
    int B, int V) {
  int i = blockIdx.x * blockDim.x + threadIdx.x;
  if (i >= B) return;

  int ui = u[i];
  int vi = v[i];

  // idx < 2^26, byte offset < 2^28 -> 32-bit offset keeps the saddr+voffset
  // global_load_b32 scale_offset form (one address VGPR per lane).
  unsigned int idx = (unsigned int)ui * (unsigned int)V + (unsigned int)vi;
  float x = logits[idx];

  // Stable f32 sigmoid: exp(-x) -> +inf => rcp = 0 ; exp(-x) -> 0 => rcp = 1.
  float e = __expf(-x);                       // v_exp_f32 (base-2 scaled)
  float s = __builtin_amdgcn_rcpf(1.0f + e);  // v_rcp_f32

  out[i] = s;
}

extern "C" void kernel_launch(void* const* d_in, const int* in_sizes, int n_in,
                              void* d_out, int out_size, void* d_ws, size_t ws_size,
                              hipStream_t stream) {
  (void)n_in; (void)out_size; (void)d_ws; (void)ws_size;

  const int*   u      = (const int*)d_in[0];   // [B] int32
  const int*   v      = (const int*)d_in[1];   // [B] int32
  const float* logits = (const float*)d_in[2]; // [V*V] float32
  float*       out    = (float*)d_out;         // [B] float32

  const int B = in_sizes[0];   // 16384
  const int V = 8192;          // reference constant; in_sizes[2] == V*V

  const int block = 256;                      // 8 wave32 waves per block
  const int grid  = (B + block - 1) / block;  // 64 blocks

  topos_yoneda_gather_sigmoid<<<grid, block, 0, stream>>>(u, v, logits, out, B, V);
}